// DifferentiableStarPlanner_68487548502378
// MI455X (gfx1250) — compile-verified
//
#include <hip/hip_runtime.h>
#include <stdint.h>

namespace {
constexpr int H = 512, W = 512;
constexpr int T = 32;            // output tile edge
constexpr int RH = 8;            // halo radius == sweeps per launch
constexpr int S = T + 2 * RH;    // 48x48 staged tile
constexpr int SWEEPS = 8;        // even => final state ends in buffer 0
constexpr int NLAUNCH = 10;      // 10 * 8 = 80 total sweeps
constexpr float OB = 10000.0f;
constexpr float FINF = 1.0e7f;
constexpr float FEPS = 1e-12f;
}

__device__ __forceinline__ int clampi(int v, int lo, int hi) {
  return v < lo ? lo : (v > hi ? hi : v);
}

__global__ __launch_bounds__(256)
void planner_init(const float* __restrict__ start, float* __restrict__ g) {
  int i = blockIdx.x * blockDim.x + threadIdx.x;
  float s = start[i];
  float v = FINF * (1.0f - s);
  g[i] = fminf(fmaxf(v, 0.0f), FINF);
}

// 8 min-plus sweeps over a 48x48 LDS-staged tile with ping-pong LDS buffers
// (one barrier per sweep); writes the exact 32x32 interior (trapezoid
// blocking: after s sweeps, points at Chebyshev distance >= s from the staged
// border are exact).
__global__ __launch_bounds__(256)
void planner_sweeps(const float* __restrict__ g_in,
                    const float* __restrict__ obs,
                    float* __restrict__ g_out) {
  __shared__ float s_g0[S * S];              // ping
  __shared__ float s_g1[S * S];              // pong
  __shared__ float s_a[S * S];               // OB_COST * obstacle, replicate-clamped
  __shared__ int s_rm[S], s_rp[S], s_cm[S], s_cp[S];  // clamped neighbor index tables

  const int tid = threadIdx.x;
  const int ty = tid >> 4, tx = tid & 15;    // 16x16 threads, 3x3 points each
  const int oy0 = (int)blockIdx.y * T - RH;
  const int ox0 = (int)blockIdx.x * T - RH;

  // Edge-replication tables (clamped to image, then to staged range; rows premul by S)
  if (tid < S) {
    int gy = oy0 + tid;
    int gx = ox0 + tid;
    s_rm[tid] = clampi(clampi(gy - 1, 0, H - 1) - oy0, 0, S - 1) * S;
    s_rp[tid] = clampi(clampi(gy + 1, 0, H - 1) - oy0, 0, S - 1) * S;
    s_cm[tid] = clampi(clampi(gx - 1, 0, W - 1) - ox0, 0, S - 1);
    s_cp[tid] = clampi(clampi(gx + 1, 0, W - 1) - ox0, 0, S - 1);
  }

  // ---- CDNA5 async global->LDS staging of g and obs tiles ----
  const unsigned lds_g = (unsigned)(uintptr_t)(void*)s_g0;
  const unsigned lds_a = (unsigned)(uintptr_t)(void*)s_a;
#pragma unroll
  for (int p = 0; p < 9; ++p) {
    const int r = ty + 16 * (p / 3);
    const int c = tx + 16 * (p % 3);
    const int gy = clampi(oy0 + r, 0, H - 1);   // replicate halo beyond image
    const int gx = clampi(ox0 + c, 0, W - 1);
    const unsigned long long ga =
        (unsigned long long)(uintptr_t)(g_in + (gy * W + gx));
    const unsigned long long oa =
        (unsigned long long)(uintptr_t)(obs + (gy * W + gx));
    const unsigned la = lds_g + 4u * (unsigned)(r * S + c);
    const unsigned lb = lds_a + 4u * (unsigned)(r * S + c);
    asm volatile("global_load_async_to_lds_b32 %0, %1, off"
                 :: "v"(la), "v"(ga) : "memory");
    asm volatile("global_load_async_to_lds_b32 %0, %1, off"
                 :: "v"(lb), "v"(oa) : "memory");
  }
  asm volatile("s_wait_asynccnt 0x0" ::: "memory");
  __syncthreads();

  // Pre-scale obstacles once: a = OB_COST * obs  (max(a_n,a_c) == OB*max(o_n,o_c) exactly)
#pragma unroll
  for (int p = 0; p < 9; ++p) {
    const int idx = (ty + 16 * (p / 3)) * S + (tx + 16 * (p % 3));
    s_a[idx] *= OB;
  }
  __syncthreads();

  // fp32 base step costs, identical to sqrt(flags + 1e-12) in float32
  const float B0 = __builtin_sqrtf(FEPS);          // sqrt(0 + eps)
  const float B1 = __builtin_sqrtf(1.0f + FEPS);   // == 1.0f
  const float B2 = __builtin_sqrtf(2.0f + FEPS);   // == sqrtf(2.0f)

  for (int s = 0; s < SWEEPS; ++s) {
    const float* __restrict__ gr = (s & 1) ? s_g1 : s_g0;  // read buffer
    float* __restrict__ gw       = (s & 1) ? s_g0 : s_g1;  // write buffer
#pragma unroll
    for (int p = 0; p < 9; ++p) {
      const int r = ty + 16 * (p / 3);
      const int c = tx + 16 * (p % 3);
      const int gy = oy0 + r;
      const int gx = ox0 + c;
      const int i0 = r * S + c;
      const bool upd = (r >= 1) && (r <= S - 2) && (c >= 1) && (c <= S - 2) &&
                       (gy >= 0) && (gy < H) && (gx >= 0) && (gx < W);
      const float gC = gr[i0];
      float best = gC;
      if (upd) {
        const int rm = s_rm[r], rp = s_rp[r], r0 = r * S;
        const int cm = s_cm[c], cp = s_cp[c];
        const int FL = (gx > 0), FR = (gx < W - 1);
        const int FU = (gy < H - 1), FD = (gy > 0);
        auto bsel = [&](int sum) { return sum == 2 ? B2 : (sum == 1 ? B1 : B0); };
        const float aC = s_a[i0];
        // obs neighbors (note reference quirk: channels 1 and 3 both use nb(-1,0))
        const float oLU = s_a[rm + cm], oU = s_a[rm + c], oRU = s_a[rm + cp];
        const float oR  = s_a[r0 + cp];
        const float oLD = s_a[rp + cm], oD = s_a[rp + c], oRD = s_a[rp + cp];
        best = fminf(best, gr[rm + cm] + (bsel(FL + FU) + fmaxf(oLU, aC)));  // c0
        best = fminf(best, gr[r0 + cm] + (bsel(FL)      + fmaxf(oU , aC)));  // c1
        best = fminf(best, gr[rp + cm] + (bsel(FL + FD) + fmaxf(oLD, aC)));  // c2
        best = fminf(best, gr[rm + c ] + (bsel(FU)      + fmaxf(oU , aC)));  // c3
        best = fminf(best, gC          + aC);                                 // c4
        best = fminf(best, gr[rp + c ] + (bsel(FD)      + fmaxf(oD , aC)));  // c5
        best = fminf(best, gr[rm + cp] + (bsel(FR + FU) + fmaxf(oRU, aC)));  // c6
        best = fminf(best, gr[r0 + cp] + (bsel(FR)      + fmaxf(oR , aC)));  // c7
        best = fminf(best, gr[rp + cp] + (bsel(FR + FD) + fmaxf(oRD, aC)));  // c8
      }
      gw[i0] = best;  // ping-pong: safe to store immediately (disjoint buffers)
    }
    __syncthreads();  // single barrier per sweep
  }

  // SWEEPS is even, so the final state is in s_g0. Write the 32x32 interior.
#pragma unroll
  for (int p = 0; p < 9; ++p) {
    const int r = ty + 16 * (p / 3);
    const int c = tx + 16 * (p % 3);
    if (r >= RH && r < RH + T && c >= RH && c < RH + T) {
      g_out[(oy0 + r) * W + (ox0 + c)] = s_g0[r * S + c];
    }
  }
}

extern "C" void kernel_launch(void* const* d_in, const int* in_sizes, int n_in,
                              void* d_out, int out_size, void* d_ws, size_t ws_size,
                              hipStream_t stream) {
  (void)in_sizes; (void)n_in; (void)out_size; (void)ws_size;
  const float* obstacles = (const float*)d_in[0];
  // d_in[1] (coords) is the meshgrid by construction; folded into constants.
  const float* start_map = (const float*)d_in[2];
  // d_in[3] (goal_map) does not affect the returned cost-to-come map.
  float* out = (float*)d_out;
  float* gA = (float*)d_ws;                     // ping
  float* gB = gA + (size_t)H * W;               // pong (2 MB total in ws)

  planner_init<<<(H * W) / 256, 256, 0, stream>>>(start_map, gA);

  dim3 grid(W / T, H / T);  // 16x16 tiles
  for (int i = 0; i < NLAUNCH; ++i) {
    const float* gin = (i & 1) ? gB : gA;
    float* gout = (i == NLAUNCH - 1) ? out : ((i & 1) ? gA : gB);
    planner_sweeps<<<grid, 256, 0, stream>>>(gin, obstacles, gout);
  }
}